// GetStoneDistAngle3DLayer_29678224015742
// MI455X (gfx1250) — compile-verified
//
#include <hip/hip_runtime.h>
#include <stdint.h>

#define S_STONES 4096
#define TPB 256
#define EPT 16   // elements (keys) per thread, sorted in registers
#define RAD2DEG 57.29577951308232087679815481410517f

// Native clang vector type: accepted by __builtin_nontemporal_store (HIP's
// float4 is a struct and is rejected).
typedef float v4f __attribute__((ext_vector_type(4)));

// Compare-exchange on u64 keys held in registers.
__device__ __forceinline__ void ce_reg(uint64_t &a, uint64_t &b, bool asc) {
    uint64_t x = a, y = b;
    bool sw = (x > y) == asc;
    a = sw ? y : x;
    b = sw ? x : y;
}

__global__ __launch_bounds__(TPB)
void GetStoneDistAngle3DLayer_kernel(const float* __restrict__ ac,
                                     const float* __restrict__ st,
                                     float* __restrict__ out) {
    __shared__ float    sStone[3 * S_STONES];   // 48 KB raw stone table (v,y,x)*4096
    __shared__ uint64_t sKeys[S_STONES];        // 32 KB sort keys

    const int row = blockIdx.x;
    const int t   = threadIdx.x;
    float* orow = out + (size_t)row * (3u * S_STONES);

    const float flag = ac[3 * row + 0];
    const float cy   = ac[3 * row + 1];
    const float cx   = ac[3 * row + 2];

    // tf.cond branch: occupied rows produce zeros. Uniform per block.
    if (flag != 0.0f) {
        v4f z = (v4f){0.f, 0.f, 0.f, 0.f};
        v4f* o4 = (v4f*)orow;
#pragma unroll
        for (int s = 0; s < (3 * S_STONES / 4) / TPB; ++s)   // 12 x 16B / thread
            __builtin_nontemporal_store(z, o4 + t + s * TPB);
        return;
    }

    // ---- Stage the 48 KB stone table into LDS with CDNA5 async global->LDS DMA.
    // GVS mode: global address is the 64-bit base in saddr plus the 32-bit offset in vaddr; LDS address is in vdst.
    {
        unsigned ldsbase = (unsigned)(uintptr_t)(&sStone[0]);
        unsigned long long gbase = (unsigned long long)(const void*)st;
#pragma unroll
        for (int s = 0; s < (3 * S_STONES * 4 / 16) / TPB; ++s) {  // 12 x B128 / thread
            unsigned byteoff = (unsigned)((t + s * TPB) * 16);
            unsigned vdst = ldsbase + byteoff;
            asm volatile("global_load_async_to_lds_b128 %0, %1, %2"
                         :: "v"(vdst), "v"(byteoff), "s"(gbase)
                         : "memory");
        }
        asm volatile("s_wait_asynccnt 0" ::: "memory");
    }
    __syncthreads();

    // ---- Phase A: build keys for the 16 owned elements directly in registers.
    const int base = t * EPT;
    uint64_t r[EPT];
#pragma unroll
    for (int e = 0; e < EPT; ++e) {
        int j = base + e;
        float dy = sStone[3 * j + 1] - cy;
        float dx = sStone[3 * j + 2] - cx;
        float d  = sqrtf(dy * dy + dx * dx);
        // dist >= 0 -> IEEE bits are monotone as u32; low bits carry index => stable sort
        r[e] = ((uint64_t)__float_as_uint(d) << 32) | (uint32_t)j;
    }

    // ---- Phase B: in-register bitonic sort of the 16-run (covers phases k=2..16).
#pragma unroll
    for (int k2 = 2; k2 <= EPT; k2 <<= 1)
#pragma unroll
        for (int j2 = k2 >> 1; j2 > 0; j2 >>= 1)
#pragma unroll
            for (int i2 = 0; i2 < EPT; ++i2)
                if ((i2 & j2) == 0)
                    ce_reg(r[i2], r[i2 | j2], (i2 & k2) == 0);
    // Bitonic precondition for phase k=32: block direction = ((base & 16) == 0).
    if (base & EPT) {
#pragma unroll
        for (int i2 = 0; i2 < EPT / 2; ++i2) {
            uint64_t tmp = r[i2]; r[i2] = r[EPT - 1 - i2]; r[EPT - 1 - i2] = tmp;
        }
    }

    // ---- Phase C: merge phases k=32..4096. Strides >=16 via LDS, strides <16 in regs.
    for (int k = 2 * EPT; k <= S_STONES; k <<= 1) {
#pragma unroll
        for (int e = 0; e < EPT; ++e) sKeys[base + e] = r[e];   // exclusive-owner write
        __syncthreads();
        for (int jj = k >> 1; jj >= EPT; jj >>= 1) {
#pragma unroll
            for (int s = 0; s < (S_STONES / 2) / TPB; ++s) {    // 8 pairs / thread
                int p = t + s * TPB;
                int i = ((p & ~(jj - 1)) << 1) | (p & (jj - 1));
                int q = i | jj;
                bool asc = (i & k) == 0;
                uint64_t a = sKeys[i];
                uint64_t b = sKeys[q];
                if ((a > b) == asc) { sKeys[i] = b; sKeys[q] = a; }  // conditional WB
            }
            __syncthreads();
        }
#pragma unroll
        for (int e = 0; e < EPT; ++e) r[e] = sKeys[base + e];   // exclusive-owner read
        bool ascb = (base & k) == 0;
#pragma unroll
        for (int j2 = EPT >> 1; j2 > 0; j2 >>= 1)
#pragma unroll
            for (int i2 = 0; i2 < EPT; ++i2)
                if ((i2 & j2) == 0)
                    ce_reg(r[i2], r[i2 | j2], ascb);
        // No barrier needed: between here and next phase's store, each LDS word is
        // touched only by its owning thread.
    }
    // After k=4096, (base & 4096)==0 for all blocks -> r[] is globally ascending.

    // ---- Phase D: gather payloads, compute angle, stream out 192 contiguous B/thread.
    float* obase = orow + base * 3;
#pragma unroll
    for (int g = 0; g < EPT / 4; ++g) {
        float f[12];
#pragma unroll
        for (int e = 0; e < 4; ++e) {
            uint64_t key = r[g * 4 + e];
            int   j    = (int)(key & 0xFFFFu);
            float dist = __uint_as_float((uint32_t)(key >> 32));
            float sv = sStone[3 * j + 0];
            float dy = sStone[3 * j + 1] - cy;
            float dx = sStone[3 * j + 2] - cx;
            float ang = atan2f(-dy, dx) * RAD2DEG;
            f[e * 3 + 0] = sv;
            f[e * 3 + 1] = dist;
            f[e * 3 + 2] = ang;
        }
        v4f* o4 = (v4f*)(obase + g * 12);   // 16B-aligned: row*49152 + t*192 + g*48
        __builtin_nontemporal_store((v4f){f[0], f[1], f[2],  f[3]},  o4 + 0);
        __builtin_nontemporal_store((v4f){f[4], f[5], f[6],  f[7]},  o4 + 1);
        __builtin_nontemporal_store((v4f){f[8], f[9], f[10], f[11]}, o4 + 2);
    }
}

extern "C" void kernel_launch(void* const* d_in, const int* in_sizes, int n_in,
                              void* d_out, int out_size, void* d_ws, size_t ws_size,
                              hipStream_t stream) {
    (void)n_in; (void)out_size; (void)d_ws; (void)ws_size;
    const float* ac = (const float*)d_in[0];   // [N,3] (flag, y, x) f32
    const float* st = (const float*)d_in[1];   // [S,3] (val, y, x) f32, S must be 4096
    float* out = (float*)d_out;                // [N, 4096, 3] f32
    const int N = in_sizes[0] / 3;
    GetStoneDistAngle3DLayer_kernel<<<dim3(N), dim3(TPB), 0, stream>>>(ac, st, out);
}